// GNNLayer_19679540150702
// MI455X (gfx1250) — compile-verified
//
#include <hip/hip_runtime.h>
#include <math.h>

#define DIM 64
#define NA  5

typedef __attribute__((ext_vector_type(2))) float v2f;
typedef __attribute__((ext_vector_type(8))) float v8f;

__device__ __forceinline__ float wave_sum32(float v) {
#pragma unroll
  for (int m = 16; m > 0; m >>= 1) v += __shfl_xor(v, m, 32);
  return v;
}

// ---------------- zero fill accumulator ----------------
__global__ __launch_bounds__(256) void zero_kernel(float* __restrict__ p, size_t n) {
  size_t i = (size_t)blockIdx.x * blockDim.x + threadIdx.x;
  if (i < n) p[i] = 0.0f;
}

// ---------------- edge kernel: one wave per edge ----------------
// lane l owns dims {2l, 2l+1}; contiguous 256B loads per gathered row.
__global__ __launch_bounds__(256) void edge_msg_kernel(
    const float* __restrict__ hidden, const float* __restrict__ rela,
    const float* __restrict__ query,  const int*   __restrict__ q_rel,
    const int*   __restrict__ edges,
    const float* __restrict__ Ws,   const float* __restrict__ Wr,
    const float* __restrict__ Wqr_w, const float* __restrict__ Wqr_b,
    const float* __restrict__ walpha_w, const float* __restrict__ walpha_b,
    float* __restrict__ agg, int E)
{
  int wave = blockIdx.x * (blockDim.x >> 5) + (threadIdx.x >> 5);
  int lane = threadIdx.x & 31;
  if (wave >= E) return;

  const int* ed = edges + (size_t)wave * 6;
  int r_idx = ed[0];
  int rel   = ed[2];
  int sub   = ed[4];
  int obj   = ed[5];
  int qr    = q_rel[r_idx];

  v2f hs = *(const v2f*)(hidden + (size_t)sub * DIM + 2 * lane);
  v2f hr = *(const v2f*)(rela   + (size_t)rel * DIM + 2 * lane);
  v2f hq = *(const v2f*)(query  + (size_t)qr  * DIM + 2 * lane);

  // pre = hs@Ws.T + hr@Wr.T + hq@Wqr_w.T + b ; alpha = sigmoid(relu(pre)@wa + ba)
  float s_alpha = walpha_b[0];
#pragma unroll
  for (int a = 0; a < NA; ++a) {
    const float* ws = Ws    + a * DIM + 2 * lane;
    const float* wr = Wr    + a * DIM + 2 * lane;
    const float* wq = Wqr_w + a * DIM + 2 * lane;
    float p = hs.x * ws[0] + hs.y * ws[1]
            + hr.x * wr[0] + hr.y * wr[1]
            + hq.x * wq[0] + hq.y * wq[1];
    p = wave_sum32(p) + Wqr_b[a];              // butterfly: all lanes see sum
    s_alpha += fmaxf(p, 0.0f) * walpha_w[a];
  }
  float alpha = 1.0f / (1.0f + expf(-s_alpha));

  const float MAXN = 0.996f;  // (1 - BALL_EPS) / sqrt(c), c = 1

  // expmap0(hs) + project
  float un = fmaxf(sqrtf(wave_sum32(hs.x * hs.x + hs.y * hs.y)), 1e-15f);
  float ts = tanhf(fminf(un, 15.0f));
  float ss = ts / un;
  float ns = fmaxf(ts, 1e-15f);
  if (ns > MAXN) ss *= MAXN / ns;
  v2f xh; xh.x = hs.x * ss; xh.y = hs.y * ss;

  // expmap0(hr) + project
  float vn = fmaxf(sqrtf(wave_sum32(hr.x * hr.x + hr.y * hr.y)), 1e-15f);
  float tr = tanhf(fminf(vn, 15.0f));
  float sr = tr / vn;
  float nr = fmaxf(tr, 1e-15f);
  if (nr > MAXN) sr *= MAXN / nr;
  v2f yh; yh.x = hr.x * sr; yh.y = hr.y * sr;

  // mobius_add
  float x2 = wave_sum32(xh.x * xh.x + xh.y * xh.y);
  float y2 = wave_sum32(yh.x * yh.x + yh.y * yh.y);
  float xy = wave_sum32(xh.x * yh.x + xh.y * yh.y);
  float cx = 1.0f + 2.0f * xy + y2;
  float cy = 1.0f - x2;
  float inv_dn = 1.0f / fmaxf(1.0f + 2.0f * xy + x2 * y2, 1e-15f);
  v2f m;
  m.x = (cx * xh.x + cy * yh.x) * inv_dn;
  m.y = (cx * xh.y + cy * yh.y) * inv_dn;

  // project + logmap0
  float mn = fmaxf(sqrtf(wave_sum32(m.x * m.x + m.y * m.y)), 1e-15f);
  float sp = (mn > MAXN) ? (MAXN / mn) : 1.0f;
  float yn = fmaxf(fminf(mn, MAXN), 1e-15f);
  float ycl = fminf(yn, 1.0f - 1e-5f);
  float at = 0.5f * (log1pf(ycl) - log1pf(-ycl));
  float msc = alpha * sp * at / yn;
  m.x *= msc; m.y *= msc;

  // segment_sum scatter (L2-resident accumulator)
  float* dst = agg + (size_t)obj * DIM + 2 * lane;
  atomicAdd(dst + 0, m.x);
  atomicAdd(dst + 1, m.y);
}

// ---------------- dense fp32 GEMM via V_WMMA_F32_16X16X4_F32 ----------------
// C[n,j] = sum_k agg[n,k] * Wh[j,k]; one wave per 16x16 tile, K-loop of 16 WMMAs.
__global__ __launch_bounds__(256) void gemm_wmma_kernel(
    const float* __restrict__ agg, const float* __restrict__ Wh,
    float* __restrict__ abuf, int N)
{
  int wave = blockIdx.x * (blockDim.x >> 5) + (threadIdx.x >> 5);
  int lane = threadIdx.x & 31;
  int rb = wave >> 2;   // 16-row block
  int cb = wave & 3;    // 16-col block (DIM/16 = 4)
  if (rb * 16 >= N) return;

  int idx   = lane & 15;            // A: row M ; B: col N (= Wh row)
  int khalf = (lane >> 4) << 1;     // lanes 0-15 -> K{0,1}, lanes 16-31 -> K{2,3}
  const float* arow = agg + (size_t)(rb * 16 + idx) * DIM + khalf;
  const float* brow = Wh  + (size_t)(cb * 16 + idx) * DIM + khalf;

  v8f acc = {};
#pragma unroll
  for (int k0 = 0; k0 < DIM; k0 += 4) {
    v2f a = *(const v2f*)(arow + k0);
    v2f b = *(const v2f*)(brow + k0);
    acc = __builtin_amdgcn_wmma_f32_16x16x4_f32(
        /*neg_a=*/false, a, /*neg_b=*/false, b,
        /*c_mod=*/(short)0, acc, /*reuse_a=*/false, /*reuse_b=*/false);
  }

  // C/D layout: VGPR j, lanes 0-15 -> M=j ; lanes 16-31 -> M=j+8 ; N = lane&15
  int row0 = rb * 16 + ((lane >> 4) << 3);
  int col  = cb * 16 + (lane & 15);
#pragma unroll
  for (int j = 0; j < 8; ++j)
    abuf[(size_t)(row0 + j) * DIM + col] = acc[j];
}

// ---------------- epilogue: p_exp_map + logmap0, one wave per row ----------------
__global__ __launch_bounds__(256) void epilogue_kernel(
    const float* __restrict__ abuf, float* __restrict__ out, int N)
{
  int wave = blockIdx.x * (blockDim.x >> 5) + (threadIdx.x >> 5);
  int lane = threadIdx.x & 31;
  if (wave >= N) return;

  v2f v = *(const v2f*)(abuf + (size_t)wave * DIM + 2 * lane);
  float nv = fmaxf(sqrtf(wave_sum32(v.x * v.x + v.y * v.y)), 1e-10f);
  float t  = tanhf(nv);
  float sc = t / nv;                      // p_exp_map
  v.x *= sc; v.y *= sc;
  float yn = fmaxf(sqrtf(wave_sum32(v.x * v.x + v.y * v.y)), 1e-15f);
  float ycl = fminf(yn, 1.0f - 1e-5f);
  float at = 0.5f * (log1pf(ycl) - log1pf(-ycl));
  float s2 = at / yn;                     // logmap0
  v2f o; o.x = v.x * s2; o.y = v.y * s2;
  *(v2f*)(out + (size_t)wave * DIM + 2 * lane) = o;
}

extern "C" void kernel_launch(void* const* d_in, const int* in_sizes, int n_in,
                              void* d_out, int out_size, void* d_ws, size_t ws_size,
                              hipStream_t stream) {
  // setup_inputs order:
  // 0 q_sub (unused), 1 q_rel, 2 hidden, 3 edges, 4 n_node, 5 old_nodes_new_idx (unused),
  // 6 rela_embed, 7 query_embed, 8 Ws, 9 Wr, 10 Wqr_w, 11 Wqr_b, 12 walpha_w, 13 walpha_b, 14 Wh
  const int*   q_rel    = (const int*)d_in[1];
  const float* hidden   = (const float*)d_in[2];
  const int*   edges    = (const int*)d_in[3];
  const float* rela     = (const float*)d_in[6];
  const float* query    = (const float*)d_in[7];
  const float* Ws       = (const float*)d_in[8];
  const float* Wr       = (const float*)d_in[9];
  const float* Wqr_w    = (const float*)d_in[10];
  const float* Wqr_b    = (const float*)d_in[11];
  const float* walpha_w = (const float*)d_in[12];
  const float* walpha_b = (const float*)d_in[13];
  const float* Wh       = (const float*)d_in[14];

  int N = in_sizes[2] / DIM;   // 100000
  int E = in_sizes[3] / 6;     // 1000000

  float* agg  = (float*)d_ws;                 // N*64 floats
  float* abuf = agg + (size_t)N * DIM;        // N*64 floats

  size_t nElem = (size_t)N * DIM;
  zero_kernel<<<(unsigned)((nElem + 255) / 256), 256, 0, stream>>>(agg, nElem);

  int wavesPerBlock = 8;  // 256 threads
  edge_msg_kernel<<<(E + wavesPerBlock - 1) / wavesPerBlock, 256, 0, stream>>>(
      hidden, rela, query, q_rel, edges, Ws, Wr, Wqr_w, Wqr_b,
      walpha_w, walpha_b, agg, E);

  int nTiles = ((N + 15) / 16) * (DIM / 16);  // 25000 waves
  gemm_wmma_kernel<<<(nTiles + wavesPerBlock - 1) / wavesPerBlock, 256, 0, stream>>>(
      agg, Wh, abuf, N);

  epilogue_kernel<<<(N + wavesPerBlock - 1) / wavesPerBlock, 256, 0, stream>>>(
      abuf, (float*)d_out, N);
}